// sobelloss_88270167868044
// MI455X (gfx1250) — compile-verified
//
#include <hip/hip_runtime.h>
#include <hip/hip_bf16.h>
#include <stdint.h>

// Problem constants (from reference: shape (16, 3, 512, 512), valid 3x3 conv)
#define IMG_H   512
#define IMG_W   512
#define N_BC    48            // 16 * 3
#define OUT_H   510
#define OUT_W   510
#define TILE_W  128           // output cols per block
#define TILE_H  32            // output rows per block
#define LD_W    132           // floats loaded per tile row = 33 x b128 chunks
#define LD_H    34            // tile rows loaded (TILE_H + 2 halo)
#define PITCH   132           // LDS row pitch in floats (132*4 = 528 B, 16B-multiple)
#define NCHUNKS (LD_H * 33)   // 1122 b128 chunks per tensor tile
#define TILES_X 4             // ceil(510/128)
#define TILES_Y 16            // ceil(510/32)
#define NBLOCKS (TILES_X * TILES_Y * N_BC)  // 3072

__device__ __forceinline__ float huber1(float d) {
    float ad = fabsf(d);
    return (ad <= 1.0f) ? 0.5f * d * d : (ad - 0.5f);
}

__global__ __launch_bounds__(256)
void sobel_huber_kernel(const float* __restrict__ inp,
                        const float* __restrict__ tgt,
                        float* __restrict__ partials) {
    __shared__ __attribute__((aligned(16))) float sA[LD_H * PITCH];
    __shared__ __attribute__((aligned(16))) float sB[LD_H * PITCH];
    __shared__ float sred[256];

    const int tid  = threadIdx.x;
    const int col0 = blockIdx.x * TILE_W;
    const int row0 = blockIdx.y * TILE_H;
    const int bc   = blockIdx.z;

    // Uniform 64-bit bases for this (b,c) image -> SGPR pair in the async load.
    const float* baseA = inp + (size_t)bc * (IMG_H * IMG_W);
    const float* baseB = tgt + (size_t)bc * (IMG_H * IMG_W);

    // ---- Stage tiles into LDS via CDNA5 async global->LDS b128 (ASYNCcnt) ----
    for (int idx = tid; idx < NCHUNKS; idx += 256) {
        const int rr = idx / 33;          // 0..33  local tile row
        const int cc = idx % 33;          // 0..32  b128 chunk within row
        const int grow = row0 + rr;       // global row
        const int gcol = col0 + cc * 4;   // global col of chunk start (mult of 4)
        // Chunks are 16B aligned and never straddle the W=512 boundary,
        // so a chunk is either fully in range or fully out.
        if (grow < IMG_H && gcol < IMG_W) {
            const unsigned voff = (unsigned)((grow * IMG_W + gcol) * 4); // byte offset
            const unsigned loff = (unsigned)((rr * PITCH + cc * 4) * 4); // LDS byte offset in tile
            // Low 32 bits of a generic pointer to LDS == LDS byte address.
            unsigned ldsA = (unsigned)(uintptr_t)(const void*)&sA[0] + loff;
            unsigned ldsB = (unsigned)(uintptr_t)(const void*)&sB[0] + loff;
            asm volatile("global_load_async_to_lds_b128 %0, %1, %2"
                         :: "v"(ldsA), "v"(voff), "s"(baseA) : "memory");
            asm volatile("global_load_async_to_lds_b128 %0, %1, %2"
                         :: "v"(ldsB), "v"(voff), "s"(baseB) : "memory");
        }
    }
    asm volatile("s_wait_asynccnt 0x0" ::: "memory");
    __syncthreads();

    // ---- Sobel + Huber: each thread = one column strip of 16 output rows ----
    const int col  = tid & 127;   // local output column 0..127
    const int rowg = tid >> 7;    // 0..1
    const int r0   = rowg * 16;   // first local output row for this thread
    const int oc   = col0 + col;  // global output column

    float acc = 0.0f;
#define DIFF(r, c) (sA[(r) * PITCH + (c)] - sB[(r) * PITCH + (c)])
    // Rolling separable Sobel: u = d[c+2]-d[c], v = d[c] + 2 d[c+1] + d[c+2]
    float u0, u1, v0, v1;
    {
        float d0 = DIFF(r0, col), d1 = DIFF(r0, col + 1), d2 = DIFF(r0, col + 2);
        u0 = d2 - d0;  v0 = d0 + 2.0f * d1 + d2;
        d0 = DIFF(r0 + 1, col); d1 = DIFF(r0 + 1, col + 1); d2 = DIFF(r0 + 1, col + 2);
        u1 = d2 - d0;  v1 = d0 + 2.0f * d1 + d2;
    }
    #pragma unroll
    for (int i = 0; i < 16; ++i) {
        const int rl = r0 + i + 2;                       // tap row (<= 33)
        float d0 = DIFF(rl, col), d1 = DIFF(rl, col + 1), d2 = DIFF(rl, col + 2);
        const float u2 = d2 - d0;
        const float v2 = d0 + 2.0f * d1 + d2;
        const int orow = row0 + r0 + i;
        if (orow < OUT_H && oc < OUT_W) {
            const float gx = u0 + 2.0f * u1 + u2;        // SOBEL_X response of (in - tg)
            const float gy = v2 - v0;                    // SOBEL_Y response of (in - tg)
            acc += huber1(gx) + huber1(gy);
        }
        u0 = u1; u1 = u2; v0 = v1; v1 = v2;
    }
#undef DIFF

    // ---- Deterministic block reduction ----
    sred[tid] = acc;
    __syncthreads();
    for (int s = 128; s > 0; s >>= 1) {
        if (tid < s) sred[tid] += sred[tid + s];
        __syncthreads();
    }
    if (tid == 0) {
        const int blk = (blockIdx.z * gridDim.y + blockIdx.y) * gridDim.x + blockIdx.x;
        partials[blk] = sred[0];
    }
}

__global__ __launch_bounds__(256)
void sobel_reduce_kernel(const float* __restrict__ partials,
                         float* __restrict__ out) {
    __shared__ float sred[256];
    const int tid = threadIdx.x;
    float a = 0.0f;
    for (int i = tid; i < NBLOCKS; i += 256) a += partials[i];
    sred[tid] = a;
    __syncthreads();
    for (int s = 128; s > 0; s >>= 1) {
        if (tid < s) sred[tid] += sred[tid + s];
        __syncthreads();
    }
    if (tid == 0) {
        const float invN = 1.0f / ((float)N_BC * (float)OUT_H * (float)OUT_W);
        out[0] = sred[0] * invN;   // mean(huber gx) + mean(huber gy), same N
    }
}

extern "C" void kernel_launch(void* const* d_in, const int* in_sizes, int n_in,
                              void* d_out, int out_size, void* d_ws, size_t ws_size,
                              hipStream_t stream) {
    (void)in_sizes; (void)n_in; (void)out_size; (void)ws_size;
    const float* inp = (const float*)d_in[0];
    const float* tgt = (const float*)d_in[1];
    float* out      = (float*)d_out;
    float* partials = (float*)d_ws;   // 3072 floats = 12 KB scratch

    dim3 grid(TILES_X, TILES_Y, N_BC);
    sobel_huber_kernel<<<grid, 256, 0, stream>>>(inp, tgt, partials);
    sobel_reduce_kernel<<<1, 256, 0, stream>>>(partials, out);
}